// RoiPoolingConv_31250182045934
// MI455X (gfx1250) — compile-verified
//
#include <hip/hip_runtime.h>
#include <stdint.h>

// ROI-align (crop_and_resize 14x14, bilinear) + 2x2 maxpool -> 7x7, 512 ROIs,
// fm = 1x38x50x512 f32 (NHWC), out = 1x512x7x7x512 f32.
//
// Store-bandwidth bound (~55 MB HBM -> ~2.4us @ 23.3 TB/s). Gather-dominated,
// no shared WMMA operand exists, so the CDNA5 paths used are:
//   - async global->LDS mover (GLOBAL_LOAD_ASYNC_TO_LDS_B128, ASYNCcnt)
//     as a double-buffered prefetch pipeline, and
//   - non-temporal output stores (TH_STORE_NT) so the write-once 51 MB output
//     streams to HBM without evicting the hot 3.9 MB feature map from L2.

#define FM_H   38
#define FM_W   50
#define NCH    512
#define CROP_N 14
#define POOL_N 7
#define IMW_F  800.0f
#define IMH_F  600.0f

#define USE_ASYNC_LDS 1

typedef float v2f __attribute__((ext_vector_type(2)));

__global__ __launch_bounds__(256) void roi_pool_kernel(
    const float* __restrict__ fm,
    const float* __restrict__ rois,
    float* __restrict__ out)
{
    // double-buffered stage: 2 bufs x 16 pixels x 512 ch = 64 KB LDS
    __shared__ __align__(16) float stage[2][16][NCH];
    __shared__ int   s_y0[CROP_N], s_y1[CROP_N], s_x0[CROP_N], s_x1[CROP_N];
    __shared__ float s_wy[CROP_N], s_wx[CROP_N];

    const int n    = blockIdx.x;
    const int t    = threadIdx.x;
    const int lane = t & 31;
    const int wave = t >> 5;

    // ---- per-ROI sample tables (reference-exact math & clipping) ----
    if (t < 2 * CROP_N) {
        const float rx1 = rois[n * 5 + 1];
        const float ry1 = rois[n * 5 + 2];
        const float rx2 = rois[n * 5 + 3];
        const float ry2 = rois[n * 5 + 4];
        if (t < CROP_N) {
            const float ty  = (float)t / (float)(CROP_N - 1);
            const float y1n = ry1 / IMH_F, y2n = ry2 / IMH_F;
            const float yy  = (y1n + ty * (y2n - y1n)) * (float)(FM_H - 1);
            const float y0f = floorf(yy);
            int y0 = (int)y0f;
            y0 = y0 < 0 ? 0 : (y0 > FM_H - 1 ? FM_H - 1 : y0);
            int y1 = y0 + 1; y1 = y1 > FM_H - 1 ? FM_H - 1 : y1;
            s_y0[t] = y0; s_y1[t] = y1; s_wy[t] = yy - y0f;
        } else {
            const int j = t - CROP_N;
            const float tx  = (float)j / (float)(CROP_N - 1);
            const float x1n = rx1 / IMW_F, x2n = rx2 / IMW_F;
            const float xx  = (x1n + tx * (x2n - x1n)) * (float)(FM_W - 1);
            const float x0f = floorf(xx);
            int x0 = (int)x0f;
            x0 = x0 < 0 ? 0 : (x0 > FM_W - 1 ? FM_W - 1 : x0);
            int x1 = x0 + 1; x1 = x1 > FM_W - 1 ? FM_W - 1 : x1;
            s_x0[j] = x0; s_x1[j] = x1; s_wx[j] = xx - x0f;
        }
    }
    __syncthreads();

    // Stage the 4x4 pixel grid needed by pooled position `pos` into buf.
    // 16 pixels x 2KB = 32KB; each of 8 waves issues exactly 8 b128 async ops,
    // so per-wave ASYNCcnt accounting below is exact.
    auto issue = [&](int pos, int buf) {
        const int pi = pos / POOL_N, pj = pos % POOL_N;
        #pragma unroll
        for (int k = 0; k < 2; ++k) {
            const int p = wave * 2 + k;          // staged pixel index 0..15
            const int a = p >> 2, b = p & 3;     // a: y-slot, b: x-slot
            const int row = 2 * pi + (a >> 1);   // crop row for this slot
            const int col = 2 * pj + (b >> 1);   // crop col for this slot
            const int ya  = (a & 1) ? s_y1[row] : s_y0[row];
            const int xb  = (b & 1) ? s_x1[col] : s_x0[col];
            const float* gbase = fm + (size_t)(ya * FM_W + xb) * NCH;
            #pragma unroll
            for (int chunk = 0; chunk < 4; ++chunk) {
                const int off = chunk * 128 + lane * 4; // floats; 16B per lane
#if USE_ASYNC_LDS
                const uint32_t laddr =
                    (uint32_t)(uintptr_t)&stage[buf][p][off];
                const uint64_t gaddr = (uint64_t)(uintptr_t)(gbase + off);
                asm volatile("global_load_async_to_lds_b128 %0, %1, off"
                             :: "v"(laddr), "v"(gaddr)
                             : "memory");
#else
                *(float4*)&stage[buf][p][off] = *(const float4*)(gbase + off);
#endif
            }
        }
    };

    const int c = t * 2;                               // 2 channels per thread
    float* outn = out + (size_t)n * POOL_N * POOL_N * NCH;

    issue(0, 0);
    for (int pos = 0; pos < POOL_N * POOL_N; ++pos) {
        if (pos + 1 < POOL_N * POOL_N) {
            issue(pos + 1, (pos + 1) & 1);
#if USE_ASYNC_LDS
            // 8 newer ops in flight; async loads complete in order per wave,
            // so ASYNCcnt<=8 => the previous batch has fully landed in LDS.
            asm volatile("s_wait_asynccnt 8" ::: "memory");
#endif
        } else {
#if USE_ASYNC_LDS
            asm volatile("s_wait_asynccnt 0" ::: "memory");
#endif
        }
        __syncthreads();  // publish buf[pos&1] to all waves

        const int buf = pos & 1;
        const int pi = pos / POOL_N, pj = pos % POOL_N;
        float m0 = -INFINITY, m1 = -INFINITY;
        #pragma unroll
        for (int di = 0; di < 2; ++di) {
            const float wy = s_wy[2 * pi + di];
            #pragma unroll
            for (int dj = 0; dj < 2; ++dj) {
                const float wx = s_wx[2 * pj + dj];
                const float2 p00 = *(const float2*)&stage[buf][(2*di+0)*4 + (2*dj+0)][c];
                const float2 p01 = *(const float2*)&stage[buf][(2*di+0)*4 + (2*dj+1)][c];
                const float2 p10 = *(const float2*)&stage[buf][(2*di+1)*4 + (2*dj+0)][c];
                const float2 p11 = *(const float2*)&stage[buf][(2*di+1)*4 + (2*dj+1)][c];
                // reference order: top/bot lerp in x, then lerp in y
                const float top0 = p00.x * (1.0f - wx) + p01.x * wx;
                const float top1 = p00.y * (1.0f - wx) + p01.y * wx;
                const float bot0 = p10.x * (1.0f - wx) + p11.x * wx;
                const float bot1 = p10.y * (1.0f - wx) + p11.y * wx;
                const float v0 = top0 * (1.0f - wy) + bot0 * wy;
                const float v1 = top1 * (1.0f - wy) + bot1 * wy;
                m0 = fmaxf(m0, v0);
                m1 = fmaxf(m1, v1);
            }
        }
        // Write-once output: stream past L2 with a non-temporal store so the
        // heavily re-gathered feature map stays L2-resident.
        v2f r; r.x = m0; r.y = m1;
        __builtin_nontemporal_store(
            r, (v2f*)&outn[(size_t)(pi * POOL_N + pj) * NCH + c]);

        __syncthreads();  // all reads of buf done before it is re-staged
    }
}

extern "C" void kernel_launch(void* const* d_in, const int* in_sizes, int n_in,
                              void* d_out, int out_size, void* d_ws, size_t ws_size,
                              hipStream_t stream) {
    const float* fm   = (const float*)d_in[0];   // 1x38x50x512 f32
    const float* rois = (const float*)d_in[1];   // Nx5 f32
    float* out        = (float*)d_out;           // 1xNx7x7x512 f32
    const int nroi = in_sizes[1] / 5;
    hipLaunchKernelGGL(roi_pool_kernel, dim3(nroi), dim3(256), 0, stream,
                       fm, rois, out);
}